// TritonMoEDispatch_10720238371207
// MI455X (gfx1250) — compile-verified
//
#include <hip/hip_runtime.h>
#include <math.h>

// Problem constants (from reference)
#define B_ 4
#define T_ 4096
#define D_ 1024
#define E_ 8
#define K_ 2
#define C_ 1537   // int(T*K/E * 1.5) + 1

typedef __attribute__((ext_vector_type(2))) float v2f;
typedef __attribute__((ext_vector_type(8))) float v8f;
typedef __attribute__((ext_vector_type(4))) int v4i;

#if __has_builtin(__builtin_amdgcn_global_load_async_to_lds_b128) && \
    __has_builtin(__builtin_amdgcn_s_wait_asynccnt)
#define USE_ASYNC_A 1
typedef __attribute__((address_space(1))) v4i* gptr_v4i;  // global (AS1) int4*
typedef __attribute__((address_space(3))) v4i* lptr_v4i;  // LDS (AS3) int4*
#endif

// ---------------------------------------------------------------------------
// Kernel 1: router logits + top-2 softmax. One wave32 per token.
// ---------------------------------------------------------------------------
__global__ __launch_bounds__(256) void moe_router(const float* __restrict__ x,
                                                  const float* __restrict__ Wr,
                                                  int* __restrict__ top_idx,
                                                  float* __restrict__ top_w) {
    __shared__ float sWr[E_ * D_];  // 32 KB, reused by 8 waves
    const int tid = threadIdx.x;
    for (int i = tid * 4; i < E_ * D_; i += 256 * 4) {
        *(float4*)&sWr[i] = *(const float4*)&Wr[i];
    }
    __syncthreads();

    const int wave = tid >> 5;
    const int lane = tid & 31;
    const long token = (long)blockIdx.x * 8 + wave;  // global token over B*T
    if (token >= (long)B_ * T_) return;

    const float* xrow = x + token * D_;
    float acc[E_];
#pragma unroll
    for (int e = 0; e < E_; ++e) acc[e] = 0.0f;

    for (int d = lane * 4; d < D_; d += 32 * 4) {
        float4 xv = *(const float4*)&xrow[d];
#pragma unroll
        for (int e = 0; e < E_; ++e) {
            float4 wv = *(const float4*)&sWr[e * D_ + d];
            acc[e] += xv.x * wv.x + xv.y * wv.y + xv.z * wv.z + xv.w * wv.w;
        }
    }
#pragma unroll
    for (int e = 0; e < E_; ++e) {
        float v = acc[e];
#pragma unroll
        for (int off = 16; off >= 1; off >>= 1) v += __shfl_xor(v, off, 32);
        acc[e] = v;
    }
    if (lane == 0) {
        float m1 = -INFINITY, m2 = -INFINITY;
        int i1 = 0, i2 = 0;
#pragma unroll
        for (int e = 0; e < E_; ++e) {
            float v = acc[e];
            if (v > m1) { m2 = m1; i2 = i1; m1 = v; i1 = e; }
            else if (v > m2) { m2 = v; i2 = e; }
        }
        float e2 = __expf(m2 - m1);
        float inv = 1.0f / (1.0f + e2);
        top_idx[token * K_ + 0] = i1;
        top_idx[token * K_ + 1] = i2;
        top_w[token * K_ + 0] = inv;
        top_w[token * K_ + 1] = e2 * inv;
    }
}

// ---------------------------------------------------------------------------
// Kernel 2: stable (t,k)-order capacity assignment per batch.
// ---------------------------------------------------------------------------
__global__ __launch_bounds__(256) void moe_scatter(const int* __restrict__ top_idx,
                                                   const float* __restrict__ top_w,
                                                   int* __restrict__ slot_token,
                                                   float* __restrict__ slot_w,
                                                   int* __restrict__ cnt) {
    const int b = blockIdx.x;
    const int tid = threadIdx.x;
    const int NS = T_ * K_;       // 8192 slots per batch
    const int CH = NS / 256;      // 32 slots per thread, contiguous (preserves order)
    const int base = b * NS + tid * CH;

    int local[E_];
#pragma unroll
    for (int e = 0; e < E_; ++e) local[e] = 0;
    for (int i = 0; i < CH; ++i) local[top_idx[base + i]]++;

    __shared__ int scnt[E_][257];
#pragma unroll
    for (int e = 0; e < E_; ++e) scnt[e][tid] = local[e];
    __syncthreads();

    if (tid < E_) {
        int run = 0;
        for (int p = 0; p < 256; ++p) { int v = scnt[tid][p]; scnt[tid][p] = run; run += v; }
        cnt[b * E_ + tid] = min(run, C_);
    }
    __syncthreads();

    int basecnt[E_];
#pragma unroll
    for (int e = 0; e < E_; ++e) basecnt[e] = scnt[e][tid];
    for (int i = 0; i < CH; ++i) {
        int slot = base + i;
        int e = top_idx[slot];
        int pos = basecnt[e]++;
        if (pos < C_) {
            int dst = (b * E_ + e) * C_ + pos;
            slot_token[dst] = (tid * CH + i) / K_;   // token index within batch
            slot_w[dst] = top_w[slot];
        }
    }
}

// ---------------------------------------------------------------------------
// Kernel 3: fused gather -> grouped GEMM (fp32 WMMA) -> weighted atomic combine.
// Block = 256 threads (8 wave32). Tile: 128 capacity rows x 64 output dims.
// Wave w computes rows [w*16, w*16+16) x 64 cols = 4 accumulators.
// sB is k-pair interleaved so each B fragment is one ds_load_b64.
// ---------------------------------------------------------------------------
#define MT 128
#define NT 64
#define KC 64
#define ASTR 68     // A row stride (floats): conflict-free fragment reads
#define BPSTR 160   // B pair-row stride (floats): 160 % 64 == 32 -> half-waves disjoint

__global__ __launch_bounds__(256) void moe_expert_gemm(const float* __restrict__ x,
                                                       const float* __restrict__ We,
                                                       const int* __restrict__ slot_token,
                                                       const float* __restrict__ slot_w,
                                                       const int* __restrict__ cnt,
                                                       float* __restrict__ out) {
    const int be = blockIdx.z;
    const int b = be / E_;
    const int e = be % E_;
    const int c0 = blockIdx.y * MT;
    const int n0 = blockIdx.x * NT;
    const int nValid = cnt[be];
    if (c0 >= nValid) return;
    const int rows = min(MT, nValid - c0);

    __shared__ float sA[MT * ASTR];          // 34.8 KB
    __shared__ float sB[(KC / 2) * BPSTR];   // 20.5 KB, [k/2][n][2] interleaved
    __shared__ int   sTok[MT];
    __shared__ float sW[MT];

    const int tid = threadIdx.x;
    if (tid < MT) {
        sTok[tid] = (tid < rows) ? slot_token[be * C_ + c0 + tid] : -1;
        sW[tid]   = (tid < rows) ? slot_w[be * C_ + c0 + tid] : 0.0f;
    }

    const int wave  = tid >> 5;
    const int lane  = tid & 31;
    const int mrow  = lane & 15;         // M (A) / N (B,C) index within 16
    const int khalf = (lane >> 4) * 2;   // K pair select: lanes 0-15 -> K0/K1, 16-31 -> K2/K3

    const v8f vzero = {0.f, 0.f, 0.f, 0.f, 0.f, 0.f, 0.f, 0.f};
    v8f accv[4];
#pragma unroll
    for (int nt = 0; nt < 4; ++nt) accv[nt] = vzero;

    const float* Wbase = We + (long)e * D_ * D_;
    const int ldr = tid >> 4;            // 0..15: rows (or k-pairs) in flight per pass
    const int ldc = (tid & 15) * 4;      // float4 column offset within 64

    // Zero over-capacity A rows once (never written inside the K loop).
    for (int z = tid; z < (MT - rows) * 16; z += 256) {
        int row = rows + (z >> 4);
        int c = (z & 15) * 4;
        *(float4*)&sA[row * ASTR + c] = make_float4(0.f, 0.f, 0.f, 0.f);
    }
    __syncthreads();                     // sTok/sW (+ tail zeros) visible

    for (int kc = 0; kc < D_; kc += KC) {
        // --- stage A: gather valid x rows (64 K floats each) via slot_token ---
#pragma unroll
        for (int p = 0; p < MT; p += 16) {
            int row = p + ldr;
            int t = sTok[row];
            if (t >= 0) {
                const float* gp = &x[((long)b * T_ + t) * D_ + kc + ldc];
                float* lp = &sA[row * ASTR + ldc];
#if defined(USE_ASYNC_A)
                __builtin_amdgcn_global_load_async_to_lds_b128((gptr_v4i)gp, (lptr_v4i)lp, 0, 0);
#else
                *(float4*)lp = *(const float4*)gp;
#endif
            }
        }
        // --- stage B: We[e][kc..kc+63][n0..n0+63], interleaved by k pairs ---
#pragma unroll
        for (int p = 0; p < KC / 2; p += 16) {
            int pr = p + ldr;                       // k-pair index 0..31
            float4 r0 = *(const float4*)&Wbase[(long)(kc + 2 * pr + 0) * D_ + n0 + ldc];
            float4 r1 = *(const float4*)&Wbase[(long)(kc + 2 * pr + 1) * D_ + n0 + ldc];
            float4 o0 = make_float4(r0.x, r1.x, r0.y, r1.y);
            float4 o1 = make_float4(r0.z, r1.z, r0.w, r1.w);
            *(float4*)&sB[pr * BPSTR + ldc * 2 + 0] = o0;
            *(float4*)&sB[pr * BPSTR + ldc * 2 + 4] = o1;
        }
#if defined(USE_ASYNC_A)
        __builtin_amdgcn_s_wait_asynccnt(0);
#endif
        __syncthreads();

        const int arow = wave * 16 + mrow;
#pragma unroll
        for (int ks = 0; ks < KC; ks += 4) {
            // A 16x4 fragment: lane holds (A[m][k0], A[m][k0+1]) -> one b64 LDS load
            v2f afrag = *(const v2f*)&sA[arow * ASTR + ks + khalf];
            const int bp = ((ks + khalf) >> 1) * BPSTR;  // k-pair row
#pragma unroll
            for (int nt = 0; nt < 4; ++nt) {
                // B 4x16 fragment: (B[k0][n], B[k0+1][n]) contiguous -> one b64 LDS load
                v2f bfrag = *(const v2f*)&sB[bp + (nt * 16 + mrow) * 2];
                accv[nt] = __builtin_amdgcn_wmma_f32_16x16x4_f32(
                    /*neg_a=*/false, afrag, /*neg_b=*/false, bfrag,
                    /*c_mod=*/(short)0, accv[nt], /*reuse_a=*/false, /*reuse_b=*/false);
            }
        }
        __syncthreads();                 // protect LDS before next staging
    }

    // --- weighted scatter-combine: out[b,t,:] += w * row ---
    // C layout: VGPR v -> M = v + (lane/16)*8, N = lane%16
#pragma unroll
    for (int nt = 0; nt < 4; ++nt) {
#pragma unroll
        for (int v = 0; v < 8; ++v) {
            int M = v + (lane >> 4) * 8;
            int row = wave * 16 + M;
            int t = sTok[row];
            if (t >= 0) {
                float val = sW[row] * accv[nt][v];
                atomicAdd(&out[((long)b * T_ + t) * D_ + n0 + nt * 16 + mrow], val);
            }
        }
    }
}

// ---------------------------------------------------------------------------
extern "C" void kernel_launch(void* const* d_in, const int* in_sizes, int n_in,
                              void* d_out, int out_size, void* d_ws, size_t ws_size,
                              hipStream_t stream) {
    const float* x  = (const float*)d_in[0];
    const float* Wr = (const float*)d_in[1];
    const float* We = (const float*)d_in[2];
    float* out = (float*)d_out;

    char* ws = (char*)d_ws;
    int*   top_idx    = (int*)ws;   ws += sizeof(int)   * (size_t)B_ * T_ * K_;
    float* top_w      = (float*)ws; ws += sizeof(float) * (size_t)B_ * T_ * K_;
    int*   slot_token = (int*)ws;   ws += sizeof(int)   * (size_t)B_ * E_ * C_;
    float* slot_w     = (float*)ws; ws += sizeof(float) * (size_t)B_ * E_ * C_;
    int*   cnt        = (int*)ws;

    (void)hipMemsetAsync(d_out, 0, (size_t)B_ * T_ * D_ * sizeof(float), stream);

    moe_router<<<dim3((B_ * T_) / 8), dim3(256), 0, stream>>>(x, Wr, top_idx, top_w);
    moe_scatter<<<dim3(B_), dim3(256), 0, stream>>>(top_idx, top_w, slot_token, slot_w, cnt);
    moe_expert_gemm<<<dim3(D_ / NT, (C_ + MT - 1) / MT, B_ * E_), dim3(256), 0, stream>>>(
        x, We, slot_token, slot_w, cnt, out);
}